// FP_27350351741230
// MI455X (gfx1250) — compile-verified
//
#include <hip/hip_runtime.h>
#include <hip/hip_bf16.h>
#include <math.h>

// Fan-beam CT forward projection for MI455X (gfx1250).
// Roofline: grid_pos (605 MB) + weighting (302 MB) streamed exactly once
//  -> ~39 us floor at 23.3 TB/s. Compute (~2 GFLOP) and image gathers
//  (576 KB image, L2/WGP$-resident) are not the bound.
// Strategy: one wave32 per ray; 64 samples per chunk; grid/weight streamed
// global->LDS with async b128/b64 ops (ASYNCcnt), 4-deep LDS pipeline
// (issue distance 3 => ~2.3 KB in flight per wave), NT temporal hint so the
// single-use stream does not evict the hot image from cache.

#define VIEW  192
#define NDET  512
#define NP    769          // samples per ray = 12*64 + 1
#define IMGW  384
#define IMGH  384
#define WPB   8            // waves per block (256 threads)
#define CHUNK 64           // samples per chunk (2 per lane)
#define NFULL 12           // full chunks (768 samples); sample 768 handled as tail
#define NBUF  4            // LDS pipeline depth

typedef unsigned long long ull;

__device__ __forceinline__ void async_b128(unsigned lds_off, ull gaddr) {
    asm volatile("global_load_async_to_lds_b128 %0, %1, off th:TH_LOAD_NT"
                 :: "v"(lds_off), "v"(gaddr) : "memory");
}
__device__ __forceinline__ void async_b64(unsigned lds_off, ull gaddr) {
    asm volatile("global_load_async_to_lds_b64 %0, %1, off th:TH_LOAD_NT"
                 :: "v"(lds_off), "v"(gaddr) : "memory");
}

// One bilinear tap (grid_sample: bilinear, zeros padding, align_corners=False),
// pre-multiplied by the ray-segment weight.
__device__ __forceinline__ float bil_tap(const float* __restrict__ img,
                                         float gx, float gy, float w) {
    const float x  = (gx + 1.0f) * (IMGW * 0.5f) - 0.5f;
    const float y  = (gy + 1.0f) * (IMGH * 0.5f) - 0.5f;
    const float xf = floorf(x);
    const float yf = floorf(y);
    const int   x0 = (int)xf;
    const int   y0 = (int)yf;
    const float wx = x - xf;
    const float wy = y - yf;

    const int x0c = min(max(x0,     0), IMGW - 1);
    const int x1c = min(max(x0 + 1, 0), IMGW - 1);
    const int y0c = min(max(y0,     0), IMGH - 1);
    const int y1c = min(max(y0 + 1, 0), IMGH - 1);

    const bool ix0 = (unsigned)(x0)     < (unsigned)IMGW;
    const bool ix1 = (unsigned)(x0 + 1) < (unsigned)IMGW;
    const bool iy0 = (unsigned)(y0)     < (unsigned)IMGH;
    const bool iy1 = (unsigned)(y0 + 1) < (unsigned)IMGH;

    const float v00 = img[y0c * IMGW + x0c];
    const float v10 = img[y0c * IMGW + x1c];
    const float v01 = img[y1c * IMGW + x0c];
    const float v11 = img[y1c * IMGW + x1c];

    const float s =
        ((1.0f - wx) * (1.0f - wy)) * ((ix0 && iy0) ? v00 : 0.0f) +
        (wx * (1.0f - wy))          * ((ix1 && iy0) ? v10 : 0.0f) +
        ((1.0f - wx) * wy)          * ((ix0 && iy1) ? v01 : 0.0f) +
        (wx * wy)                   * ((ix1 && iy1) ? v11 : 0.0f);
    return w * s;
}

__global__ __launch_bounds__(WPB * 32)
void fp_ct_fanbeam_kernel(const float* __restrict__ img,   // [384*384]
                          const float* __restrict__ grid,  // [V*D*P*2]
                          const float* __restrict__ wgt,   // [V*D*P]
                          float* __restrict__ out) {       // [V*D]
    // chunk layout in LDS: lane i owns samples (2i, 2i+1) of the chunk
    __shared__ float4 g_sh[WPB][NBUF][32];   // (x0,y0,x1,y1) per lane
    __shared__ float2 w_sh[WPB][NBUF][32];   // (w0,w1) per lane

    const int lane = threadIdx.x & 31;
    const int wave = threadIdx.x >> 5;
    const int ray  = blockIdx.x * WPB + wave;        // < VIEW*NDET
    const size_t base = (size_t)ray * NP;            // element base into wgt

    // per-lane byte addresses for chunk 0
    const ull Ggrid = (ull)(size_t)(grid + base * 2) + (ull)lane * 16;
    const ull Gwgt  = (ull)(size_t)(wgt  + base)     + (ull)lane * 8;
    const unsigned Lg = (unsigned)(size_t)&g_sh[wave][0][lane]; // buf stride 512 B
    const unsigned Lw = (unsigned)(size_t)&w_sh[wave][0][lane]; // buf stride 256 B

    // ---- prologue: fill 3 of 4 pipeline stages (6 async ops in flight) ----
    #pragma unroll
    for (int k = 0; k < NBUF - 1; ++k) {
        async_b128(Lg + (unsigned)k * 512u, Ggrid + (ull)k * 512u);
        async_b64 (Lw + (unsigned)k * 256u, Gwgt  + (ull)k * 256u);
    }

    float acc = 0.0f;

    // ---- main loop: chunks 0..8, issuing chunk c+3, oldest completes at <=6 ----
    for (int c = 0; c < NFULL - (NBUF - 1); ++c) {
        const int nb = (c + NBUF - 1) & (NBUF - 1);      // buffer for chunk c+3
        asm volatile("s_wait_dscnt 0x0" ::: "memory");   // WAR: prior DS reads done
        async_b128(Lg + (unsigned)nb * 512u, Ggrid + (ull)(c + NBUF - 1) * 512u);
        async_b64 (Lw + (unsigned)nb * 256u, Gwgt  + (ull)(c + NBUF - 1) * 256u);
        asm volatile("s_wait_asynccnt 0x6" ::: "memory"); // chunk c has landed

        const int buf = c & (NBUF - 1);
        const float4 g = g_sh[wave][buf][lane];
        const float2 w = w_sh[wave][buf][lane];
        acc += bil_tap(img, g.x, g.y, w.x);
        acc += bil_tap(img, g.z, g.w, w.y);
    }

    // ---- peeled drain: chunks 9, 10, 11 with waits 4, 2, 0 ----
    {
        asm volatile("s_wait_asynccnt 0x4" ::: "memory");
        const float4 g = g_sh[wave][9 & (NBUF - 1)][lane];
        const float2 w = w_sh[wave][9 & (NBUF - 1)][lane];
        acc += bil_tap(img, g.x, g.y, w.x);
        acc += bil_tap(img, g.z, g.w, w.y);
    }
    {
        asm volatile("s_wait_asynccnt 0x2" ::: "memory");
        const float4 g = g_sh[wave][10 & (NBUF - 1)][lane];
        const float2 w = w_sh[wave][10 & (NBUF - 1)][lane];
        acc += bil_tap(img, g.x, g.y, w.x);
        acc += bil_tap(img, g.z, g.w, w.y);
    }
    {
        asm volatile("s_wait_asynccnt 0x0" ::: "memory");
        const float4 g = g_sh[wave][11 & (NBUF - 1)][lane];
        const float2 w = w_sh[wave][11 & (NBUF - 1)][lane];
        acc += bil_tap(img, g.x, g.y, w.x);
        acc += bil_tap(img, g.z, g.w, w.y);
    }

    // ---- tail: single sample p = 768, lane 0 only (plain global loads) ----
    if (lane == 0) {
        const float gx = grid[(base + (NP - 1)) * 2 + 0];
        const float gy = grid[(base + (NP - 1)) * 2 + 1];
        const float w  = wgt[base + (NP - 1)];
        acc += bil_tap(img, gx, gy, w);
    }

    // ---- wave32 reduction ----
    #pragma unroll
    for (int off = 16; off > 0; off >>= 1)
        acc += __shfl_xor(acc, off, 32);

    if (lane == 0)
        out[ray] = (acc != acc) ? 0.0f : acc;   // NaN -> 0 (reference semantics)
}

extern "C" void kernel_launch(void* const* d_in, const int* in_sizes, int n_in,
                              void* d_out, int out_size, void* d_ws, size_t ws_size,
                              hipStream_t stream) {
    (void)in_sizes; (void)n_in; (void)d_ws; (void)ws_size; (void)out_size;
    const float* img  = (const float*)d_in[0];   // [1,1,384,384]
    const float* grid = (const float*)d_in[1];   // [192,512,769,2]
    const float* wgt  = (const float*)d_in[2];   // [1,1,192,512,769]
    float* out = (float*)d_out;                  // [1,1,192,512] = 98304

    const int rays   = VIEW * NDET;              // 98304
    const int blocks = rays / WPB;               // 12288
    fp_ct_fanbeam_kernel<<<blocks, WPB * 32, 0, stream>>>(img, grid, wgt, out);
}